// QuantumCircuit_65240553226490
// MI455X (gfx1250) — compile-verified
//
#include <hip/hip_runtime.h>
#include <math.h>

// ---------------------------------------------------------------------------
// 18-qubit state-vector circuit, fused into 4 full-state sweeps + 2 tiny
// kernels. Qubit w <-> flat bit (17-w). Split: low 13 bits local (LDS tiles),
// high 5 bits handled in registers (32 elements / thread).
// ---------------------------------------------------------------------------

#define NQ    18
#define DIM   (1 << NQ)        // 262144
#define LOWB  13
#define LOWN  (1 << LOWB)      // 8192 elements per tile
#define HIGHN 32               // 2^5 tiles
#define IS2   0.70710678118654752440f

// workspace layout (float offsets); total = 788480 floats ~= 3.01 MiB
#define WS_INV  0
#define WS_CRY  64                 // 18*18 CRY angle table [t*18+c]
#define WS_CRZ  448                // 18*18 CRZ angle table [tgt*18+ctrl]
#define WS_PART 832                // 32*18 Z partials
#define WS_REAL 2048               // real state after phase A (DIM floats)
#define WS_CPLX (2048 + DIM)       // complex state (2*DIM floats)

typedef __attribute__((ext_vector_type(2))) float v2f;
typedef __attribute__((ext_vector_type(8))) float v8f;

// ---- CDNA5 async global -> LDS copy (ASYNCcnt tracked) --------------------
__device__ __forceinline__ void async_b128(void* lds_dst, const void* gsrc) {
  unsigned l = (unsigned)(uintptr_t)lds_dst;           // low 32 bits = LDS addr
  unsigned long long g = (unsigned long long)(uintptr_t)gsrc;
  asm volatile("global_load_async_to_lds_b128 %0, %1, off"
               :: "v"(l), "v"(g) : "memory");
}
__device__ __forceinline__ void async_wait0() {
  asm volatile("s_wait_asynccnt 0" ::: "memory");
}

// ---------------------------------------------------------------------------
// Kernel 0: inv-norm + angle tables
// ---------------------------------------------------------------------------
__global__ __launch_bounds__(512) void k_prep(const float* __restrict__ in,
                                              const float* __restrict__ conn,
                                              const float* __restrict__ th,
                                              const float* __restrict__ ph,
                                              float* __restrict__ ws) {
  __shared__ float red[512];
  const int tid = threadIdx.x;
  float s = 0.f;
  for (int i = tid; i < DIM; i += 512) { float v = in[i]; s += v * v; }
  red[tid] = s;
  __syncthreads();
  for (int off = 256; off > 0; off >>= 1) {
    if (tid < off) red[tid] += red[tid + off];
    __syncthreads();
  }
  if (tid == 0) ws[WS_INV] = 1.0f / sqrtf(red[0]);

  if (tid < 324) {
    const int a = tid / 18, b = tid % 18;
    if (b < a) {                       // CRY: control c=b < target t=a
      const int c = b, t = a;
      const int idx = 18 + c * 17 - (c * (c - 1)) / 2 + (t - c - 1);
      const float cv = conn[c * 18 + t];
      ws[WS_CRY + t * 18 + c] = (cv != 0.0f) ? th[idx] : 0.0f;
    } else if (b > a) {                // CRZ: target tgt=a < control ctrl=b
      const int tgt = a, ctrl = b;
      const int pidx = tgt * 17 - (tgt * (tgt - 1)) / 2 + (ctrl - tgt - 1);
      const float cv = conn[ctrl * 18 + tgt];
      ws[WS_CRZ + tgt * 18 + ctrl] =
          (cv != 0.0f) ? (ph[pidx] * 0.9f + 0.1f * cv) : 0.0f;
    }
  }
}

// ---------------------------------------------------------------------------
// Kernel 1: normalize + fused RY/CRY target-groups t=0..4 (flat bits 17..13),
// 32 strided elements per thread, fully in registers.
// ---------------------------------------------------------------------------
__global__ __launch_bounds__(256) void k_a1(const float* __restrict__ in,
                                            const float* __restrict__ th,
                                            float* __restrict__ ws) {
  const int tid = blockIdx.x * 256 + threadIdx.x;   // 13-bit low index
  const float inv = ws[WS_INV];
  const float* cry = ws + WS_CRY;
  float v[32];
#pragma unroll
  for (int k = 0; k < 32; k++) v[k] = in[tid + (k << LOWB)] * inv;
#pragma unroll
  for (int t = 0; t <= 4; t++) {
    const int kb = 4 - t;                           // target bit within k
    const float base = th[t];
#pragma unroll
    for (int k0 = 0; k0 < 32; k0++) {
      if (k0 & (1 << kb)) continue;
      const int k1 = k0 | (1 << kb);
      float ang = base;
#pragma unroll
      for (int c = 0; c < t; c++)                   // controls: k bits 4-c
        if (k0 & (1 << (4 - c))) ang += cry[t * 18 + c];
      float sn, cs;
      __sincosf(0.5f * ang, &sn, &cs);
      const float a = v[k0], b = v[k1];
      v[k0] = cs * a - sn * b;
      v[k1] = sn * a + cs * b;
    }
  }
#pragma unroll
  for (int k = 0; k < 32; k++) ws[WS_REAL + tid + (k << LOWB)] = v[k];
}

// ---------------------------------------------------------------------------
// Kernel 2: fused RY/CRY target-groups t=5..17 (flat bits 12..0) in a 32 KB
// LDS tile; global control bits (tile id) fold into a scalar base angle.
// ---------------------------------------------------------------------------
__global__ __launch_bounds__(256) void k_a2(const float* __restrict__ th,
                                            float* __restrict__ ws) {
  __shared__ __align__(16) float tile[LOWN];        // 32 KB
  const int g = blockIdx.x, tid = threadIdx.x;
  const float* cry = ws + WS_CRY;
  float* st = ws + WS_REAL + (g << LOWB);
  for (int i = tid; i < LOWN / 4; i += 256)
    async_b128(&tile[i * 4], st + i * 4);
  async_wait0();
  __syncthreads();
  for (int t = 5; t <= 17; t++) {
    const int p = 17 - t;                           // local target bit
    float base = th[t];
    for (int c = 0; c < 5; c++)                     // global controls
      if (g & (1 << (4 - c))) base += cry[t * 18 + c];
    for (int q = tid; q < LOWN / 2; q += 256) {
      const int j0 = ((q >> p) << (p + 1)) | (q & ((1 << p) - 1));
      const int j1 = j0 | (1 << p);
      float ang = base;
      for (int c = 5; c < t; c++)                   // local controls
        if (j0 & (1 << (17 - c))) ang += cry[t * 18 + c];
      float sn, cs;
      __sincosf(0.5f * ang, &sn, &cs);
      const float a = tile[j0], b = tile[j1];
      tile[j0] = cs * a - sn * b;
      tile[j1] = sn * a + cs * b;
    }
    __syncthreads();
  }
  for (int i = tid; i < LOWN; i += 256) st[i] = tile[i];
}

// ---------------------------------------------------------------------------
// Kernel 3: QFT groups tgt=0..4 (H + fused CRZ phases), real -> complex,
// 32 elements / thread in registers.
// ---------------------------------------------------------------------------
__global__ __launch_bounds__(256) void k_b1(float* __restrict__ ws) {
  const int tid = blockIdx.x * 256 + threadIdx.x;   // 13-bit low index
  const float* crz = ws + WS_CRZ;
  float re[32], im[32];
#pragma unroll
  for (int k = 0; k < 32; k++) {
    re[k] = ws[WS_REAL + tid + (k << LOWB)];
    im[k] = 0.f;
  }
  float fix[5];                                     // tid-bit phase parts
#pragma unroll
  for (int tgt = 0; tgt < 5; tgt++) {
    float s = 0.f;
#pragma unroll
    for (int ctrl = 5; ctrl < 18; ctrl++)
      if (tid & (1 << (17 - ctrl))) s += crz[tgt * 18 + ctrl];
    fix[tgt] = s;
  }
#pragma unroll
  for (int tgt = 0; tgt < 5; tgt++) {
    const int kb = 4 - tgt;
#pragma unroll
    for (int k0 = 0; k0 < 32; k0++) {               // H butterfly
      if (k0 & (1 << kb)) continue;
      const int k1 = k0 | (1 << kb);
      const float ar = re[k0], ai = im[k0], br = re[k1], bi = im[k1];
      re[k0] = (ar + br) * IS2; im[k0] = (ai + bi) * IS2;
      re[k1] = (ar - br) * IS2; im[k1] = (ai - bi) * IS2;
    }
#pragma unroll
    for (int k = 0; k < 32; k++) {                  // fused diagonal phase
      float phi = fix[tgt];
#pragma unroll
      for (int ctrl = tgt + 1; ctrl < 5; ctrl++)
        if (k & (1 << (4 - ctrl))) phi += crz[tgt * 18 + ctrl];
      phi *= 0.5f;
      float sn, cs;
      __sincosf(phi, &sn, &cs);
      if (!(k & (1 << kb))) sn = -sn;               // tgt bit 0 -> e^{-i phi}
      const float a = re[k], b = im[k];
      re[k] = a * cs - b * sn;
      im[k] = a * sn + b * cs;
    }
  }
  float2* cst = (float2*)(ws + WS_CPLX);
#pragma unroll
  for (int k = 0; k < 32; k++)
    cst[tid + (k << LOWB)] = make_float2(re[k], im[k]);
}

// ---------------------------------------------------------------------------
// Kernel 4: QFT groups tgt=5..17 in a 64 KB complex LDS tile (320 KB WGP LDS),
// plus fused |amp|^2 Z-expectation partials (state never written back).
// ---------------------------------------------------------------------------
__global__ __launch_bounds__(256) void k_b2(float* __restrict__ ws) {
  __shared__ __align__(16) float2 tile[LOWN];       // 64 KB
  const int g = blockIdx.x, tid = threadIdx.x;
  const float* crz = ws + WS_CRZ;
  const float2* src = (const float2*)(ws + WS_CPLX) + (g << LOWB);
  for (int i = tid; i < LOWN / 2; i += 256)
    async_b128(&tile[i * 2], src + i * 2);
  async_wait0();
  __syncthreads();
  for (int tgt = 5; tgt <= 17; tgt++) {
    const int p = 17 - tgt;
    for (int q = tid; q < LOWN / 2; q += 256) {
      const int j0 = ((q >> p) << (p + 1)) | (q & ((1 << p) - 1));
      const int j1 = j0 | (1 << p);
      const float2 A = tile[j0], B = tile[j1];
      const float ur = (A.x + B.x) * IS2, ui = (A.y + B.y) * IS2;
      const float wr = (A.x - B.x) * IS2, wi = (A.y - B.y) * IS2;
      float phi = 0.f;                              // controls all below bit p
      for (int ctrl = tgt + 1; ctrl < 18; ctrl++)
        if (j0 & (1 << (17 - ctrl))) phi += crz[tgt * 18 + ctrl];
      phi *= 0.5f;
      float sn, cs;
      __sincosf(phi, &sn, &cs);
      tile[j0] = make_float2(ur * cs + ui * sn, -ur * sn + ui * cs); // e^{-i}
      tile[j1] = make_float2(wr * cs - wi * sn,  wr * sn + wi * cs); // e^{+i}
    }
    __syncthreads();
  }
  // Z expectation partials: z[w] += (bit(17-w)==0 ? +p : -p)
  float acc[NQ];
#pragma unroll
  for (int w = 0; w < NQ; w++) acc[w] = 0.f;
  for (int i = tid; i < LOWN; i += 256) {
    const float2 A = tile[i];
    const float pr = A.x * A.x + A.y * A.y;
    const int x = (g << LOWB) | i;
#pragma unroll
    for (int w = 0; w < NQ; w++)
      acc[w] += (x & (1 << (17 - w))) ? -pr : pr;
  }
  __syncthreads();
  float* red = (float*)tile;                        // reuse tile as scratch
#pragma unroll
  for (int w = 0; w < NQ; w++) red[w * 256 + tid] = acc[w];
  __syncthreads();
  for (int off = 128; off > 0; off >>= 1) {
    if (tid < off) {
#pragma unroll
      for (int w = 0; w < NQ; w++)
        red[w * 256 + tid] += red[w * 256 + tid + off];
    }
    __syncthreads();
  }
  if (tid < NQ) ws[WS_PART + g * NQ + tid] = red[tid * 256];
}

// ---------------------------------------------------------------------------
// Kernel 5: final reduction 32x18 -> 18 via V_WMMA_F32_16X16X4_F32.
//   Fold 32 tiles -> Q[k][w] (k=0..3) on VALU, then D = A x B with
//   A[m][k] = Q[k][m] (documented 16x4 layout: lane = M + 16*(K>=2),
//   VGPR = K&1) and B = all-ones (layout-insensitive):
//   D[m][n] = sum_k Q[k][m] = out[m] for every n.
//   D layout: lane 0 VGPR r = out[r]; lane 16 VGPR r = out[8+r].
//   One full wave32, uniform control through the WMMA (EXEC all 1s).
// ---------------------------------------------------------------------------
__global__ __launch_bounds__(32) void k_final(const float* __restrict__ ws,
                                              float* __restrict__ out) {
#if __has_builtin(__builtin_amdgcn_wmma_f32_16x16x4_f32)
  const int t = threadIdx.x;
  const int m = t & 15;
  const int kb = (t >> 4) * 2;                      // K of this lane's VGPR0
  float a0 = 0.f, a1 = 0.f;
#pragma unroll
  for (int j = 0; j < 8; j++) {
    a0 += ws[WS_PART + ((kb    ) * 8 + j) * NQ + m];   // Q[kb][m]
    a1 += ws[WS_PART + ((kb + 1) * 8 + j) * NQ + m];   // Q[kb+1][m]
  }
  v2f A; A.x = a0; A.y = a1;
  v2f B; B.x = 1.0f; B.y = 1.0f;
  v8f C = {};
  C = __builtin_amdgcn_wmma_f32_16x16x4_f32(false, A, false, B, (short)0, C,
                                            false, false);
  if (t == 0) {
#pragma unroll
    for (int r = 0; r < 8; r++) out[r] = C[r];      // out[0..7]
    float s16 = 0.f, s17 = 0.f;
    for (int g = 0; g < HIGHN; g++) {
      s16 += ws[WS_PART + g * NQ + 16];
      s17 += ws[WS_PART + g * NQ + 17];
    }
    out[16] = s16;
    out[17] = s17;
  } else if (t == 16) {
#pragma unroll
    for (int r = 0; r < 8; r++) out[8 + r] = C[r];  // out[8..15]
  }
#else
  const int w = threadIdx.x;
  if (w < NQ) {
    float s = 0.f;
    for (int g = 0; g < HIGHN; g++) s += ws[WS_PART + g * NQ + w];
    out[w] = s;
  }
#endif
}

extern "C" void kernel_launch(void* const* d_in, const int* in_sizes, int n_in,
                              void* d_out, int out_size, void* d_ws, size_t ws_size,
                              hipStream_t stream) {
  (void)in_sizes; (void)n_in; (void)out_size; (void)ws_size;
  const float* in   = (const float*)d_in[0];   // input_state   [262144]
  const float* conn = (const float*)d_in[1];   // connections   [18*18]
  const float* th   = (const float*)d_in[2];   // ry_thetas     [171]
  const float* ph   = (const float*)d_in[3];   // qft_phases    [153]
  float* ws  = (float*)d_ws;                   // needs >= ~3.01 MiB
  float* out = (float*)d_out;

  k_prep <<<1,     512, 0, stream>>>(in, conn, th, ph, ws);
  k_a1   <<<HIGHN, 256, 0, stream>>>(in, th, ws);
  k_a2   <<<HIGHN, 256, 0, stream>>>(th, ws);
  k_b1   <<<HIGHN, 256, 0, stream>>>(ws);
  k_b2   <<<HIGHN, 256, 0, stream>>>(ws);
  k_final<<<1,      32, 0, stream>>>(ws, out);
}